// DQN_52201032515996
// MI455X (gfx1250) — compile-verified
//
#include <hip/hip_runtime.h>
#include <hip/hip_bf16.h>
#include <math.h>

// ---------------------------------------------------------------------------
// Problem constants (from the reference)
// ---------------------------------------------------------------------------
#define HIDDEN      3
#define LAYERS      120
#define INPUT_LAYER 262144
#define OUTPUTS     18

#define NBLK 256          // blocks for the big matvec
#define NTHR 256          // threads per block

// Padded LDS tiles so WMMA operand loads are unconditional ds_load_b64
#define A_ROWS 16         // M padding for A (valid rows: 0..2)
#define B_ROWS 32         // N padding for B (valid rows: 0..17)

typedef __attribute__((ext_vector_type(2))) float v2f;
typedef __attribute__((ext_vector_type(8))) float v8f;

// ---------------------------------------------------------------------------
// Kernel 1: y[r] = sum_k W_ih0[r][k] * x[k]   (bandwidth bound, 4 MB traffic)
// 65536 threads, one float4 of x and three float4s of W per thread;
// deterministic fixed-tree block reduction -> per-block partials in d_ws.
// ---------------------------------------------------------------------------
__global__ void k_dot(const float* __restrict__ x,
                      const float* __restrict__ W0,
                      float* __restrict__ partials) {
    __shared__ float red0[NTHR];
    __shared__ float red1[NTHR];
    __shared__ float red2[NTHR];

    const int t   = threadIdx.x;
    const int gid = blockIdx.x * NTHR + t;          // 0 .. 65535 (float4 index)

    const float4* __restrict__ x4 = (const float4*)x;
    const float4* __restrict__ w4 = (const float4*)W0;
    const int row4 = INPUT_LAYER / 4;               // 65536 float4 per W row

    const float4 xv = x4[gid];

    float s0, s1, s2;
    {
        float4 w = w4[0 * row4 + gid];
        s0 = fmaf(w.x, xv.x, fmaf(w.y, xv.y, fmaf(w.z, xv.z, w.w * xv.w)));
        w = w4[1 * row4 + gid];
        s1 = fmaf(w.x, xv.x, fmaf(w.y, xv.y, fmaf(w.z, xv.z, w.w * xv.w)));
        w = w4[2 * row4 + gid];
        s2 = fmaf(w.x, xv.x, fmaf(w.y, xv.y, fmaf(w.z, xv.z, w.w * xv.w)));
    }

    red0[t] = s0; red1[t] = s1; red2[t] = s2;
    __syncthreads();
    for (int off = NTHR / 2; off > 0; off >>= 1) {
        if (t < off) {
            red0[t] += red0[t + off];
            red1[t] += red1[t + off];
            red2[t] += red2[t + off];
        }
        __syncthreads();
    }
    if (t == 0) {
        partials[blockIdx.x * 3 + 0] = red0[0];
        partials[blockIdx.x * 3 + 1] = red1[0];
        partials[blockIdx.x * 3 + 2] = red2[0];
    }
}

// ---------------------------------------------------------------------------
// Kernel 2 (single block):
//   A) deterministic reduction of the 256 per-block partials -> y[3]
//   B) h0 = tanh(y + b_ih0 + b_hh0); sequential 119-layer 3x3 chain (lane 0),
//      writing H^T (padded to 16 x 120) into LDS
//   C) head GEMM  out(3x18) = H^T @ W_head^T + b_head  via
//      V_WMMA_F32_16X16X4_F32 on wave 0 (K=120 -> 30 steps, 2 N-tiles),
//      all operand reads are unconditional 8B-aligned ds_load_b64.
// ---------------------------------------------------------------------------
__global__ void k_chain_head(const float* __restrict__ partials,
                             const float* __restrict__ b_ih0,
                             const float* __restrict__ b_hh0,
                             const float* __restrict__ W_ih,
                             const float* __restrict__ b_ih,
                             const float* __restrict__ b_hh,
                             const float* __restrict__ W_head,
                             const float* __restrict__ b_head,
                             float* __restrict__ out) {
    __shared__ float red0[NTHR];
    __shared__ float red1[NTHR];
    __shared__ float red2[NTHR];
    // HT[m][k] = H[k][m]  (A-matrix, rows m>=3 zero)
    __shared__ __align__(16) float HT[A_ROWS * LAYERS];
    // WT[o][k] = W_head[o][k] (B-matrix source, rows o>=18 zero)
    __shared__ __align__(16) float WT[B_ROWS * LAYERS];

    const int t = threadIdx.x;

    // ---- A) reduce partials (fixed tree -> bitwise deterministic) ----
    red0[t] = partials[t * 3 + 0];
    red1[t] = partials[t * 3 + 1];
    red2[t] = partials[t * 3 + 2];
    __syncthreads();
    for (int off = NTHR / 2; off > 0; off >>= 1) {
        if (t < off) {
            red0[t] += red0[t + off];
            red1[t] += red1[t + off];
            red2[t] += red2[t + off];
        }
        __syncthreads();
    }

    // ---- zero A padding rows (m = 3..15) ----
    for (int i = t; i < (A_ROWS - HIDDEN) * LAYERS; i += NTHR)
        HT[HIDDEN * LAYERS + i] = 0.0f;
    // ---- stage W_head and zero B padding rows (o = 18..31) ----
    for (int i = t; i < OUTPUTS * LAYERS; i += NTHR)
        WT[i] = W_head[i];
    for (int i = t; i < (B_ROWS - OUTPUTS) * LAYERS; i += NTHR)
        WT[OUTPUTS * LAYERS + i] = 0.0f;

    // ---- B) sequential RNN chain on a single lane ----
    if (t == 0) {
        float h[HIDDEN];
        const float y[HIDDEN] = { red0[0], red1[0], red2[0] };
        for (int r = 0; r < HIDDEN; ++r) {
            h[r] = tanhf(y[r] + b_ih0[r] + b_hh0[r]);
            HT[r * LAYERS + 0] = h[r];
        }
        for (int l = 1; l < LAYERS; ++l) {
            const float* W  = W_ih + (l - 1) * HIDDEN * HIDDEN;
            const float* bi = b_ih + (l - 1) * HIDDEN;
            const float* bh = b_hh + (l - 1) * HIDDEN;
            float nh[HIDDEN];
            for (int r = 0; r < HIDDEN; ++r) {
                float acc = bi[r] + bh[r];
                for (int c = 0; c < HIDDEN; ++c)
                    acc = fmaf(W[r * HIDDEN + c], h[c], acc);
                nh[r] = tanhf(acc);
            }
            for (int r = 0; r < HIDDEN; ++r) {
                h[r] = nh[r];
                HT[r * LAYERS + l] = h[r];
            }
        }
    }
    __syncthreads();

    // ---- C) head via f32 WMMA (wave 0 only; EXEC all-1s inside branch) ----
    if (t < 32) {
        const int half = t >> 4;                 // 0: lanes 0-15, 1: lanes 16-31
        const int col  = t & 15;

        // Per-lane operand base pointers; ka = k0 + 2*half is even, and
        // col*120*4 % 8 == 0, so all v2f loads are 8B-aligned ds_load_b64.
        const float* __restrict__ Ab  = HT + col * LAYERS;          // A row m=col
        const float* __restrict__ B0b = WT + col * LAYERS;          // o = col
        const float* __restrict__ B1b = WT + (16 + col) * LAYERS;   // o = 16+col

        v8f c0 = {};                             // outputs o = 0..15
        v8f c1 = {};                             // outputs o = 16..17

        for (int k0 = 0; k0 < LAYERS; k0 += 4) {
            const int ka = k0 + 2 * half;        // K for .x ; .y is ka+1
            const v2f a  = *(const v2f*)(Ab  + ka);
            const v2f b0 = *(const v2f*)(B0b + ka);
            const v2f b1 = *(const v2f*)(B1b + ka);
            // 8 args: (neg_a, A, neg_b, B, c_mod, C, reuse_a, reuse_b)
            c0 = __builtin_amdgcn_wmma_f32_16x16x4_f32(
                     false, a, false, b0, (short)0, c0, false, false);
            c1 = __builtin_amdgcn_wmma_f32_16x16x4_f32(
                     false, a, false, b1, (short)0, c1, false, false);
        }

        // D layout: VGPR r on lanes 0-15 holds row M=r, N=col.
        if (half == 0) {
            for (int r = 0; r < HIDDEN; ++r) {
                out[r * OUTPUTS + col] = c0[r] + b_head[col];
                if (col < OUTPUTS - 16)
                    out[r * OUTPUTS + 16 + col] = c1[r] + b_head[16 + col];
            }
        }
    }
}

// ---------------------------------------------------------------------------
// Launch
// ---------------------------------------------------------------------------
extern "C" void kernel_launch(void* const* d_in, const int* in_sizes, int n_in,
                              void* d_out, int out_size, void* d_ws, size_t ws_size,
                              hipStream_t stream) {
    const float* x      = (const float*)d_in[0];
    const float* W_ih0  = (const float*)d_in[1];
    const float* b_ih0  = (const float*)d_in[2];
    const float* b_hh0  = (const float*)d_in[3];
    const float* W_ih   = (const float*)d_in[4];
    const float* b_ih   = (const float*)d_in[5];
    const float* b_hh   = (const float*)d_in[6];
    const float* W_head = (const float*)d_in[7];
    const float* b_head = (const float*)d_in[8];
    float*       out    = (float*)d_out;
    float*       part   = (float*)d_ws;              // NBLK*3 floats

    k_dot<<<NBLK, NTHR, 0, stream>>>(x, W_ih0, part);
    k_chain_head<<<1, NTHR, 0, stream>>>(part, b_ih0, b_hh0,
                                         W_ih, b_ih, b_hh,
                                         W_head, b_head, out);
}